// doubleObjectiveWeightedCrossEntropy_38517266710750
// MI455X (gfx1250) — compile-verified
//
#include <hip/hip_runtime.h>
#include <hip/hip_bf16.h>
#include <stdint.h>

#define CLS 20

typedef float fvec4 __attribute__((ext_vector_type(4)));

// ---- CDNA5 async global -> LDS copy (ASYNCcnt path), portable via inline asm ----
__device__ __forceinline__ void async_load_b32_to_lds(unsigned lds_byte_off, const void* gptr) {
    unsigned long long ga = (unsigned long long)(uintptr_t)gptr;
    // vdst = VGPR holding LDS byte address, vaddr = 64-bit global address VGPR pair
    asm volatile("global_load_async_to_lds_b32 %0, %1, off"
                 :: "v"(lds_byte_off), "v"(ga)
                 : "memory");
}

__device__ __forceinline__ void wait_asynccnt0() {
    asm volatile("s_wait_asynccnt 0" ::: "memory");
}

// Low 32 bits of a generic pointer to __shared__ data = LDS byte offset (LDS aperture
// lives entirely in addr[63:32]).
__device__ __forceinline__ unsigned lds_off_of(const void* p) {
    return (unsigned)(uintptr_t)p;
}

__global__ __launch_bounds__(256) void ce_loss_partial_kernel(
    const float* __restrict__ pred,   // [N, 20]
    const float* __restrict__ area,   // [N]
    const float* __restrict__ cw,     // [20] class_weights
    const float* __restrict__ cm,     // [20] class_modes
    const float* __restrict__ bt,     // [20] bottom_thresholds
    const float* __restrict__ mw,     // [20] max_weights
    const int*  __restrict__ tgt,     // [N]
    float* __restrict__ partial,      // [gridDim.x]
    int n)
{
    __shared__ float s_tab[4 * CLS];  // [cw | cm | bt | mw]
    __shared__ float s_wsum[8];

    const int tid = threadIdx.x;

    // Stage the 4 tiny per-class tables into LDS with CDNA5 async loads.
    if (tid < 4 * CLS) {
        const int tb = tid / CLS;
        const int el = tid - tb * CLS;
        const float* src = (tb == 0) ? cw : (tb == 1) ? cm : (tb == 2) ? bt : mw;
        async_load_b32_to_lds(lds_off_of(&s_tab[tid]), src + el);
    }
    wait_asynccnt0();     // per-wave; waves that issued wait for their async copies
    __syncthreads();      // publish LDS table to all waves in the block

    float acc = 0.0f;
    const int stride = gridDim.x * blockDim.x;
    for (int i = blockIdx.x * blockDim.x + tid; i < n; i += stride) {
        // Row of 20 floats = 80 B, 16B-aligned -> 5 non-temporal b128 loads.
        const fvec4* r4 = (const fvec4*)(pred + (size_t)i * CLS);
        fvec4 p0 = __builtin_nontemporal_load(r4 + 0);
        fvec4 p1 = __builtin_nontemporal_load(r4 + 1);
        fvec4 p2 = __builtin_nontemporal_load(r4 + 2);
        fvec4 p3 = __builtin_nontemporal_load(r4 + 3);
        fvec4 p4 = __builtin_nontemporal_load(r4 + 4);
        float x[CLS] = {p0.x, p0.y, p0.z, p0.w,
                        p1.x, p1.y, p1.z, p1.w,
                        p2.x, p2.y, p2.z, p2.w,
                        p3.x, p3.y, p3.z, p3.w,
                        p4.x, p4.y, p4.z, p4.w};
        int   t = __builtin_nontemporal_load(&tgt[i]);
        float a = __builtin_nontemporal_load(&area[i]);

        // Stable logsumexp; select x[t] branch-free (no dynamic reg indexing).
        float m = x[0];
#pragma unroll
        for (int c = 1; c < CLS; ++c) m = fmaxf(m, x[c]);
        float s = 0.0f, xt = 0.0f;
#pragma unroll
        for (int c = 0; c < CLS; ++c) {
            s += __expf(x[c] - m);
            xt = (c == t) ? x[c] : xt;
        }
        const float lse = m + __logf(s);
        const float nll = lse - xt;

        const float wc   = s_tab[t];
        const float mode = s_tab[CLS + t];
        const float th   = s_tab[2 * CLS + t];
        const float mwt  = s_tab[3 * CLS + t];
        const float interp = mwt - (a - th) / (mode - th) * (mwt - 1.0f);
        const float sw = (a < mode) ? ((a < th) ? mwt : interp) : 1.0f;

        acc += nll * wc * sw;
    }

    // wave32 xor-shuffle reduction, then cross-wave via LDS.
#pragma unroll
    for (int off = 16; off > 0; off >>= 1) acc += __shfl_xor(acc, off, 32);
    if ((tid & 31) == 0) s_wsum[tid >> 5] = acc;
    __syncthreads();
    if (tid < 8) {
        float w = s_wsum[tid];
#pragma unroll
        for (int off = 4; off > 0; off >>= 1) w += __shfl_xor(w, off, 32);
        if (tid == 0) partial[blockIdx.x] = w;
    }
}

__global__ __launch_bounds__(256) void ce_loss_final_kernel(
    const float* __restrict__ partial, int nparts,
    float* __restrict__ out, float inv_n)
{
    __shared__ float s_wsum[8];
    const int tid = threadIdx.x;
    float acc = 0.0f;
    for (int i = tid; i < nparts; i += blockDim.x) acc += partial[i];
#pragma unroll
    for (int off = 16; off > 0; off >>= 1) acc += __shfl_xor(acc, off, 32);
    if ((tid & 31) == 0) s_wsum[tid >> 5] = acc;
    __syncthreads();
    if (tid < 8) {
        float w = s_wsum[tid];
#pragma unroll
        for (int off = 4; off > 0; off >>= 1) w += __shfl_xor(w, off, 32);
        if (tid == 0) out[0] = w * inv_n;
    }
}

extern "C" void kernel_launch(void* const* d_in, const int* in_sizes, int n_in,
                              void* d_out, int out_size, void* d_ws, size_t ws_size,
                              hipStream_t stream) {
    const float* pred = (const float*)d_in[0];
    const float* area = (const float*)d_in[1];
    const float* cw   = (const float*)d_in[2];
    const float* cm   = (const float*)d_in[3];
    const float* bt   = (const float*)d_in[4];
    const float* mw   = (const float*)d_in[5];
    const int*   tgt  = (const int*)d_in[6];

    const int n = in_sizes[1];            // N (area element count)
    float* partial = (float*)d_ws;        // 2048 floats of scratch

    const int NB = 2048;                  // 2048 blocks x 8 wave32 = deep occupancy
    ce_loss_partial_kernel<<<NB, 256, 0, stream>>>(pred, area, cw, cm, bt, mw, tgt,
                                                   partial, n);
    ce_loss_final_kernel<<<1, 256, 0, stream>>>(partial, NB, (float*)d_out,
                                                1.0f / (float)n);
}